// HEModule_10290741641713
// MI455X (gfx1250) — compile-verified
//
#include <hip/hip_runtime.h>
#include <hip/hip_bf16.h>

typedef __attribute__((ext_vector_type(2))) float v2f;
typedef __attribute__((ext_vector_type(4))) float v4f;
typedef __attribute__((ext_vector_type(8))) float v8f;

#define NUM_BINS 256
#define SLICES   48            // B*C = 16*3
#define PIX      (1024*1024)   // H*W per slice
#define HIST_BLOCKS_PER_SLICE 64
#define MAP_BLOCKS_PER_SLICE  128

__device__ __forceinline__ int binof(float v) {
    v = fminf(fmaxf(v, 0.0f), 1.0f);
    int q = (int)(v * 255.0f);          // truncation, matches astype(int32)
    q = q < 0 ? 0 : q;
    return q > 255 ? 255 : q;
}

// ---------------- Pass 0: zero the histogram workspace ----------------
__global__ void he_zero_kernel(unsigned int* __restrict__ hist) {
    hist[blockIdx.x * NUM_BINS + threadIdx.x] = 0u;
}

// ---------------- Pass 1: per-slice histograms (LDS -> L2 atomics) ----------------
// x loaded with default RT hints so its lines stay in L2 for pass 3's re-read.
__global__ void he_hist_kernel(const float* __restrict__ x,
                               unsigned int* __restrict__ hist) {
    __shared__ unsigned int lh[NUM_BINS];
    const int slice = blockIdx.y;
    const int chunk = blockIdx.x;
    const int tid   = threadIdx.x;      // 0..255

    lh[tid] = 0u;
    __syncthreads();

    const int pixPerBlock = PIX / HIST_BLOCKS_PER_SLICE;   // 16384
    const v4f* __restrict__ p =
        (const v4f*)(x + (size_t)slice * PIX + (size_t)chunk * pixPerBlock);
    const int vecs = pixPerBlock / 4;                      // 4096

    for (int v = tid; v < vecs; v += 256) {
        if (v + 512 < vecs) __builtin_prefetch(&p[v + 512], 0, 1); // global_prefetch_b8
        v4f f = p[v];
        atomicAdd(&lh[binof(f.x)], 1u);
        atomicAdd(&lh[binof(f.y)], 1u);
        atomicAdd(&lh[binof(f.z)], 1u);
        atomicAdd(&lh[binof(f.w)], 1u);
    }
    __syncthreads();

    unsigned int c = lh[tid];
    if (c) atomicAdd(&hist[slice * NUM_BINS + tid], c);
}

// ---------------- Pass 2: cumsum(256) + normalize via WMMA ----------------
// One wave (32 threads) per channel. hist reshaped row-major to 16x16 Hm.
// P = Hm @ T  (T[r][c] = r<=c)  => inclusive in-row prefix, computed with
// 4x V_WMMA_F32_16X16X4_F32 accumulating over K. P[m][15] is the row total,
// so cross-row offsets are a tiny LDS reduction. f32 exact for counts < 2^24.
__global__ void he_cdf_kernel(const unsigned int* __restrict__ hist,
                              float* __restrict__ cdf) {
    __shared__ float h[NUM_BINS];
    __shared__ float rowtot[16];

    const int ch   = blockIdx.x;
    const int lane = threadIdx.x;       // 0..31, wave32
    const int m    = lane & 15;         // A-frag row == B-frag column (n)
    const int hi   = lane >> 4;         // lane-half selector

    const unsigned int* __restrict__ hc = hist + ch * NUM_BINS;
    #pragma unroll
    for (int i = 0; i < 8; ++i) h[lane + 32 * i] = (float)hc[lane + 32 * i];
    __syncthreads();

    v8f P = {};                         // C/D accumulator, zero
    #pragma unroll
    for (int k = 0; k < 4; ++k) {
        // A-frag of Hm (16x4 slice, K columns 4k..4k+3):
        //   a[v] holds A[m][4k + v + 2*hi]  (ISA 32-bit A layout)
        v2f a;
        a.x = h[m * 16 + 4 * k + 2 * hi + 0];
        a.y = h[m * 16 + 4 * k + 2 * hi + 1];
        // B-frag of T (4x16 slice): b[v] holds T[4k + v + 2*hi][n], n = lane&15
        const int r0 = 4 * k + 2 * hi;
        v2f b;
        b.x = (r0     <= m) ? 1.0f : 0.0f;
        b.y = (r0 + 1 <= m) ? 1.0f : 0.0f;
        P = __builtin_amdgcn_wmma_f32_16x16x4_f32(
                /*neg_a=*/false, a, /*neg_b=*/false, b,
                /*c_mod=*/(short)0, P, /*reuse_a=*/false, /*reuse_b=*/false);
    }

    // D layout: VGPR r of lane -> element (row = r + 8*hi, col = lane&15).
    // Column 15 carries the full row sums.
    if (m == 15) {
        #pragma unroll
        for (int r = 0; r < 8; ++r) rowtot[r + 8 * hi] = P[r];
    }
    __syncthreads();

    float denom = 0.0f;
    #pragma unroll
    for (int q = 0; q < 16; ++q) denom += rowtot[q];
    denom = fmaxf(denom, 1.0f);

    float* __restrict__ out = cdf + ch * NUM_BINS;
    #pragma unroll
    for (int r = 0; r < 8; ++r) {
        const int mm = r + 8 * hi;
        float off = 0.0f;
        for (int q = 0; q < mm; ++q) off += rowtot[q];
        out[mm * 16 + m] = (P[r] + off) / denom;
    }
}

// ---------------- Pass 3: remap through per-channel CDF ----------------
// x is read for the LAST time -> NT load; y is never re-read -> NT store.
// Keeps L2 free to serve pass-1-resident x lines.
__global__ void he_remap_kernel(const float* __restrict__ x,
                                const float* __restrict__ cdf,
                                float* __restrict__ y) {
    __shared__ float lc[NUM_BINS];
    const int slice = blockIdx.y;
    const int chunk = blockIdx.x;
    const int tid   = threadIdx.x;      // 0..255

    lc[tid] = cdf[slice * NUM_BINS + tid];
    __syncthreads();

    const int pixPerBlock = PIX / MAP_BLOCKS_PER_SLICE;    // 8192
    const size_t base = (size_t)slice * PIX + (size_t)chunk * pixPerBlock;
    const v4f* __restrict__ xp = (const v4f*)(x + base);
    v4f*       __restrict__ yp = (v4f*)(y + base);
    const int vecs = pixPerBlock / 4;                      // 2048

    for (int v = tid; v < vecs; v += 256) {
        v4f f = __builtin_nontemporal_load(xp + v);
        v4f o;
        o.x = lc[binof(f.x)];
        o.y = lc[binof(f.y)];
        o.z = lc[binof(f.z)];
        o.w = lc[binof(f.w)];
        __builtin_nontemporal_store(o, yp + v);
    }
}

extern "C" void kernel_launch(void* const* d_in, const int* in_sizes, int n_in,
                              void* d_out, int out_size, void* d_ws, size_t ws_size,
                              hipStream_t stream) {
    (void)in_sizes; (void)n_in; (void)out_size; (void)ws_size;
    const float* x = (const float*)d_in[0];
    float*       y = (float*)d_out;

    unsigned int* hist = (unsigned int*)d_ws;                              // 48*256 u32
    float*        cdf  = (float*)((char*)d_ws + SLICES * NUM_BINS * sizeof(unsigned int));

    he_zero_kernel <<<dim3(SLICES),                    dim3(NUM_BINS), 0, stream>>>(hist);
    he_hist_kernel <<<dim3(HIST_BLOCKS_PER_SLICE, SLICES), dim3(256),  0, stream>>>(x, hist);
    he_cdf_kernel  <<<dim3(SLICES),                    dim3(32),       0, stream>>>(hist, cdf);
    he_remap_kernel<<<dim3(MAP_BLOCKS_PER_SLICE, SLICES),  dim3(256),  0, stream>>>(x, cdf, y);
}